// FeedForwardQuantum_65481071396389
// MI455X (gfx1250) — compile-verified
//
#include <hip/hip_runtime.h>

typedef __attribute__((ext_vector_type(2))) float v2f;
typedef __attribute__((ext_vector_type(4))) float v4f;
typedef __attribute__((ext_vector_type(8))) float v8f;

#define EMBED 1024
#define TOK_PER_WAVE 16
#define WAVES_PER_BLOCK 8
#define TOK_PER_BLOCK (TOK_PER_WAVE * WAVES_PER_BLOCK)

// D = A(16x4 f32) * B(4x16 f32) + C, all-fp32 WMMA used as a 16-token GEMV
// reduction engine: B's 16 columns all hold W1, so every accumulator column
// equals the theta vector after the K loop.
__global__ __launch_bounds__(256) void ffq_wmma_kernel(
    const float* __restrict__ x,   // [32768, 1024]
    const float* __restrict__ W1,  // [1024]
    const float* __restrict__ b1,  // [1]
    const float* __restrict__ phi, // [1]
    const float* __restrict__ W2,  // [1024]
    const float* __restrict__ b2,  // [1024]
    float* __restrict__ out,       // [32768, 1024]
    int tokens)
{
    __shared__ float sW1[EMBED];

    const int tid = threadIdx.x;
    // Stage W1 (4 KB) into LDS once per block: 256 threads x float4.
    *(v4f*)(sW1 + tid * 4) = *(const v4f*)(W1 + tid * 4);
    __syncthreads();

    const int lane  = tid & 31;
    const int wave  = tid >> 5;
    const long t0   = ((long)blockIdx.x * WAVES_PER_BLOCK + wave) * TOK_PER_WAVE;
    if (t0 >= tokens) return;              // wave-uniform: EXEC stays all-ones

    const int half8 = (lane >> 4) * 8;     // lanes 0-15 -> cols k..k+7, 16-31 -> k+8..k+15
    const int mrow  = lane & 15;           // token row within tile
    const float* xrow = x + (t0 + mrow) * EMBED;

    // Preload W2/b2 slices into registers; lane covers d = j*128 + lane*4 .. +3.
    v4f w2r[8], b2r[8];
#pragma unroll
    for (int j = 0; j < 8; ++j) {
        w2r[j] = *(const v4f*)(W2 + j * 128 + lane * 4);
        b2r[j] = *(const v4f*)(b2 + j * 128 + lane * 4);
    }

    v8f acc = {};
    // K loop: 64 chunks of 16 columns; per chunk 2 coalesced global b128 loads,
    // 2 LDS b128 loads, 4 fp32 WMMAs.
#pragma unroll 4
    for (int k = 0; k < EMBED; k += 16) {
        const int cb = k + half8;
        v4f a0 = *(const v4f*)(xrow + cb);
        v4f a1 = *(const v4f*)(xrow + cb + 4);
        v4f w0 = *(const v4f*)(sW1 + cb);
        v4f w1 = *(const v4f*)(sW1 + cb + 4);

        v2f aa, bb;
        aa[0] = a0[0]; aa[1] = a0[1]; bb[0] = w0[0]; bb[1] = w0[1];
        acc = __builtin_amdgcn_wmma_f32_16x16x4_f32(false, aa, false, bb, (short)0, acc, false, false);
        aa[0] = a0[2]; aa[1] = a0[3]; bb[0] = w0[2]; bb[1] = w0[3];
        acc = __builtin_amdgcn_wmma_f32_16x16x4_f32(false, aa, false, bb, (short)0, acc, false, false);
        aa[0] = a1[0]; aa[1] = a1[1]; bb[0] = w1[0]; bb[1] = w1[1];
        acc = __builtin_amdgcn_wmma_f32_16x16x4_f32(false, aa, false, bb, (short)0, acc, false, false);
        aa[0] = a1[2]; aa[1] = a1[3]; bb[0] = w1[2]; bb[1] = w1[3];
        acc = __builtin_amdgcn_wmma_f32_16x16x4_f32(false, aa, false, bb, (short)0, acc, false, false);
    }

    const float b1v  = b1[0];
    const float cphi = __cosf(phi[0]);

    // Accumulator layout: VGPR j holds theta for token j (lanes 0-15) and
    // token j+8 (lanes 16-31); every lane of each half holds the same value.
#pragma unroll
    for (int m = 0; m < 16; ++m) {
        float th = __shfl(acc[m & 7], (m >> 3) << 4, 32) + b1v;
        float qm = __cosf(th) * cphi;
        float* orow = out + (t0 + m) * EMBED;
#pragma unroll
        for (int j = 0; j < 8; ++j) {
            v4f o = qm * w2r[j] + b2r[j];   // fma4 from registers
            *(v4f*)(orow + j * 128 + lane * 4) = o;  // 512 B contiguous per wave store
        }
    }
}

extern "C" void kernel_launch(void* const* d_in, const int* in_sizes, int n_in,
                              void* d_out, int out_size, void* d_ws, size_t ws_size,
                              hipStream_t stream) {
    const float* x   = (const float*)d_in[0];
    const float* W1  = (const float*)d_in[1];
    const float* b1  = (const float*)d_in[2];
    const float* phi = (const float*)d_in[3];
    const float* W2  = (const float*)d_in[4];
    const float* b2  = (const float*)d_in[5];
    float* out = (float*)d_out;

    const int tokens = in_sizes[0] / EMBED;                 // 8*4096 = 32768
    const int blocks = (tokens + TOK_PER_BLOCK - 1) / TOK_PER_BLOCK;  // 256

    ffq_wmma_kernel<<<blocks, 256, 0, stream>>>(x, W1, b1, phi, W2, b2, out, tokens);
}